// Mamba_53730040873453
// MI455X (gfx1250) — compile-verified
//
#include <hip/hip_runtime.h>
#include <math.h>

// ---- problem dims (match reference) ----
#define BSZ       32
#define TLEN      1024
#define OBS       128
#define HID       128
#define D_STATE   32
#define D_CONVW   4
#define NHEADS    2
#define HEADDIM   64
#define D_INNER   128          // NHEADS*HEADDIM
#define D_CONV_CH 192          // D_INNER + 2*D_STATE
#define D_IN_PROJ 322          // 2*D_INNER + 2*D_STATE + NHEADS
#define ZX_LD     336          // padded row stride for zxbcdt (multiple of 16)
#define N_ACT     8
#define MTOT      (BSZ * TLEN) // 32768 rows

typedef __attribute__((ext_vector_type(2))) float v2f;
typedef __attribute__((ext_vector_type(8))) float v8f;

__device__ __forceinline__ float gelu_exact(float v) {
    // jax.nn.gelu(approximate=False): 0.5*x*(1+erf(x/sqrt(2)))
    return 0.5f * v * (1.0f + erff(v * 0.70710678118654752f));
}

// ---------------------------------------------------------------------------
// GEMM: C[M,N] = act(A[M,ldA] @ W[N,K]^T + bias)
// One wave32 per 16x16 output tile, V_WMMA_F32_16X16X4_F32, K stepped by 4.
// f32 WMMA VGPR layout (ISA 7.12.2):
//   A 16x4 : lane l -> m = l&15, k = 2*(l>>4) + {0,1}
//   B 4x16 : lane l -> n = l&15, k = 2*(l>>4) + {0,1}
//   C 16x16: lane l, vgpr r -> m = r + 8*(l>>4), n = l&15
// ---------------------------------------------------------------------------
template <int ACT>
__global__ void gemm_nt_wmma(const float* __restrict__ A, int ldA,
                             const float* __restrict__ W,   // N x K row-major
                             const float* __restrict__ bias,
                             float* __restrict__ C, int ldC,
                             int N, int K) {
    const int lane = threadIdx.x & 31;
    const int m0 = blockIdx.x * 16;
    const int n0 = blockIdx.y * 16;
    const int li = lane & 15;
    const int kh = (lane >> 4) * 2;

    const int mrow = m0 + li;
    const int ncol = n0 + li;
    const float* arow = A + (size_t)mrow * ldA;
    const float* wrow = W + (size_t)(ncol < N ? ncol : 0) * K;  // clamp: column never stored

    v8f acc = {};
    for (int k = 0; k < K; k += 4) {
        v2f a, b;
        a.x = arow[k + kh];
        a.y = arow[k + kh + 1];
        b.x = wrow[k + kh];
        b.y = wrow[k + kh + 1];
        acc = __builtin_amdgcn_wmma_f32_16x16x4_f32(
            /*neg_a=*/false, a, /*neg_b=*/false, b,
            /*c_mod=*/(short)0, acc, /*reuse_a=*/false, /*reuse_b=*/false);
    }

    if (ncol < N) {
        const float bv = bias ? bias[ncol] : 0.0f;
        const int mbase = m0 + ((lane >> 4) << 3);
#pragma unroll
        for (int r = 0; r < 8; ++r) {
            float v = acc[r] + bv;
            if (ACT == 1) v = gelu_exact(v);
            C[(size_t)(mbase + r) * ldC + ncol] = v;
        }
    }
}

// ---------------------------------------------------------------------------
// Depthwise causal conv(4) + SiLU on xBC channels; softplus(dt)+dA per head.
// zx layout per row (stride ZX_LD): [0:128)=z  [128:320)=xBC_raw  [320:322)=dt
// ---------------------------------------------------------------------------
__global__ void conv_dt_kernel(const float* __restrict__ zx,
                               const float* __restrict__ conv_w,  // [192,4]
                               const float* __restrict__ conv_b,  // [192]
                               const float* __restrict__ dt_bias, // [2]
                               const float* __restrict__ A_log,   // [2]
                               float* __restrict__ xbc,           // [MTOT,192]
                               float* __restrict__ dtb,           // [MTOT,2]
                               float* __restrict__ dab) {         // [MTOT,2]
    const int bt = blockIdx.x;
    const int t  = bt & (TLEN - 1);
    const int c  = threadIdx.x;
    if (c < D_CONV_CH) {
        float acc = conv_b[c];
#pragma unroll
        for (int k = 0; k < D_CONVW; ++k) {
            const int tt = t + k - (D_CONVW - 1);
            if (tt >= 0)
                acc += zx[(size_t)(bt + k - (D_CONVW - 1)) * ZX_LD + D_INNER + c] *
                       conv_w[c * D_CONVW + k];
        }
        xbc[(size_t)bt * D_CONV_CH + c] = acc / (1.0f + expf(-acc));  // silu
    } else if (c < D_CONV_CH + NHEADS) {
        const int h = c - D_CONV_CH;
        const float x = zx[(size_t)bt * ZX_LD + (D_IN_PROJ - NHEADS) + h] + dt_bias[h];
        const float dt = (x > 20.0f) ? x : log1pf(expf(x));  // softplus
        dtb[bt * NHEADS + h] = dt;
        dab[bt * NHEADS + h] = expf(dt * -expf(A_log[h]));
    }
}

// ---------------------------------------------------------------------------
// Sequential SSM scan + D-skip + SiLU(z) gate + RMSNorm, fused.
// One block per batch; 128 threads = 128 inner channels; 32-entry state row
// per thread in registers. B_t/C_t staged in LDS; RMSNorm reduction across
// the 128 threads of the block.
// ---------------------------------------------------------------------------
__global__ void scan_kernel(const float* __restrict__ zx,
                            const float* __restrict__ xbc,
                            const float* __restrict__ dtb,
                            const float* __restrict__ dab,
                            const float* __restrict__ Dp,
                            const float* __restrict__ norm_w,
                            float* __restrict__ ybuf) {   // [MTOT,128]
    const int b = blockIdx.x;
    const int c = threadIdx.x;      // 0..127 : channel = h*64 + p
    const int hh = c >> 6;

    float S[D_STATE];
#pragma unroll
    for (int n = 0; n < D_STATE; ++n) S[n] = 0.0f;

    __shared__ float sBC[2 * D_STATE];   // B_t then C_t
    __shared__ float red[D_INNER];

    const float Dh = Dp[hh];
    const float nw = norm_w[c];

    for (int t = 0; t < TLEN; ++t) {
        const size_t bt = (size_t)b * TLEN + t;
        if (c < 2 * D_STATE) sBC[c] = xbc[bt * D_CONV_CH + D_INNER + c];
        __syncthreads();

        const float dA  = dab[bt * NHEADS + hh];
        const float dt  = dtb[bt * NHEADS + hh];
        const float xh  = xbc[bt * D_CONV_CH + c];
        const float dtx = dt * xh;

        float y = 0.0f;
#pragma unroll
        for (int n = 0; n < D_STATE; ++n) {
            S[n] = S[n] * dA + dtx * sBC[n];
            y += S[n] * sBC[D_STATE + n];
        }
        y += Dh * xh;

        const float z = zx[bt * ZX_LD + c];
        y *= z / (1.0f + expf(-z));      // * silu(z)

        red[c] = y * y;
        __syncthreads();
        for (int s = D_INNER / 2; s > 0; s >>= 1) {
            if (c < s) red[c] += red[c + s];
            __syncthreads();
        }
        const float scale = rsqrtf(red[0] * (1.0f / D_INNER) + 1e-5f);
        ybuf[bt * D_INNER + c] = y * scale * nw;
        __syncthreads();
    }
}

// ---------------------------------------------------------------------------
// Decoder: logits[bt,8] = h@dec_w^T + dec_b ; values[bt] = h.val_w + val_b
// d_out = [logits (MTOT*8) | values (MTOT)]
// ---------------------------------------------------------------------------
__global__ void dec_kernel(const float* __restrict__ outh,
                           const float* __restrict__ dec_w,
                           const float* __restrict__ dec_b,
                           const float* __restrict__ val_w,
                           const float* __restrict__ val_b,
                           float* __restrict__ out) {
    const int idx = blockIdx.x * blockDim.x + threadIdx.x;
    if (idx >= MTOT * (N_ACT + 1)) return;
    const int bt = idx / (N_ACT + 1);
    const int j  = idx - bt * (N_ACT + 1);
    const float* w = (j < N_ACT) ? (dec_w + (size_t)j * HID) : val_w;
    float s = (j < N_ACT) ? dec_b[j] : val_b[0];
    const float* hrow = outh + (size_t)bt * HID;
#pragma unroll 4
    for (int k = 0; k < HID; ++k) s += hrow[k] * w[k];
    if (j < N_ACT) out[(size_t)bt * N_ACT + j] = s;
    else           out[(size_t)MTOT * N_ACT + bt] = s;
}

// ---------------------------------------------------------------------------
extern "C" void kernel_launch(void* const* d_in, const int* in_sizes, int n_in,
                              void* d_out, int out_size, void* d_ws, size_t ws_size,
                              hipStream_t stream) {
    const float* x          = (const float*)d_in[0];
    const float* enc_w      = (const float*)d_in[1];
    const float* enc_b      = (const float*)d_in[2];
    const float* in_proj_w  = (const float*)d_in[3];
    const float* conv_w     = (const float*)d_in[4];
    const float* conv_b     = (const float*)d_in[5];
    const float* dt_bias    = (const float*)d_in[6];
    const float* A_log      = (const float*)d_in[7];
    const float* Dp         = (const float*)d_in[8];
    const float* norm_w     = (const float*)d_in[9];
    const float* out_proj_w = (const float*)d_in[10];
    const float* dec_w      = (const float*)d_in[11];
    const float* dec_b      = (const float*)d_in[12];
    const float* val_w      = (const float*)d_in[13];
    const float* val_b      = (const float*)d_in[14];

    float* ws  = (float*)d_ws;
    float* zx  = ws;                                  // MTOT * 336
    float* h   = zx  + (size_t)MTOT * ZX_LD;          // MTOT * 128 (reused as ybuf)
    float* xbc = h   + (size_t)MTOT * HID;            // MTOT * 192 (reused as outh)
    float* dtb = xbc + (size_t)MTOT * D_CONV_CH;      // MTOT * 2
    float* dab = dtb + (size_t)MTOT * NHEADS;         // MTOT * 2
    float* ybuf = h;        // h dead after in_proj
    float* outh = xbc;      // xbc dead after scan
    float* out  = (float*)d_out;

    const dim3 wblk(32);  // one wave32 per block = one 16x16 WMMA tile

    // 1) encoder: h = gelu(x @ enc_w^T + enc_b)
    gemm_nt_wmma<1><<<dim3(MTOT / 16, HID / 16), wblk, 0, stream>>>(
        x, OBS, enc_w, enc_b, h, HID, HID, OBS);

    // 2) in_proj: zx = h @ in_proj_w^T   (N=322, guarded tiles)
    gemm_nt_wmma<0><<<dim3(MTOT / 16, (D_IN_PROJ + 15) / 16), wblk, 0, stream>>>(
        h, HID, in_proj_w, nullptr, zx, ZX_LD, D_IN_PROJ, HID);

    // 3) causal depthwise conv + silu; softplus(dt) and dA
    conv_dt_kernel<<<MTOT, 256, 0, stream>>>(zx, conv_w, conv_b, dt_bias, A_log,
                                             xbc, dtb, dab);

    // 4) fused SSM scan + gate + RMSNorm
    scan_kernel<<<BSZ, D_INNER, 0, stream>>>(zx, xbc, dtb, dab, Dp, norm_w, ybuf);

    // 5) out_proj: outh = ybuf @ out_proj_w^T
    gemm_nt_wmma<0><<<dim3(MTOT / 16, HID / 16), wblk, 0, stream>>>(
        ybuf, HID, out_proj_w, nullptr, outh, HID, HID, HID);

    // 6) decoder heads -> d_out
    const int tot = MTOT * (N_ACT + 1);
    dec_kernel<<<(tot + 255) / 256, 256, 0, stream>>>(outh, dec_w, dec_b,
                                                      val_w, val_b, out);
}